// RankOrderCoding_32521492365351
// MI455X (gfx1250) — compile-verified
//
#include <hip/hip_runtime.h>
#include <stdint.h>
#include <stddef.h>

// Problem constants (from reference): data [8, 512, 512] f32, T=32.
#define RB   8
#define RN   (512 * 512)   // 262144 elements per batch
#define RT   32
#define TOPK 31            // ranks 0..30 are individual; rank>=31 -> t=31
#define NBLK 64            // select-blocks per batch
#define CHUNK (RN / NBLK)  // 4096 elements per select-block
#define NTHREADS 256
#define CAND 32            // candidates kept per select-block
#define FILL_IT 8          // float4 stores per thread in fill kernel

typedef __attribute__((ext_vector_type(4))) float f4;
typedef int v4i __attribute__((vector_size(16)));
typedef unsigned long long u64;
typedef unsigned int u32;

// CDNA5 async global->LDS copy path (ASYNCcnt), guarded so compile never breaks.
#if defined(__gfx1250__) && \
    __has_builtin(__builtin_amdgcn_global_load_async_to_lds_b128) && \
    __has_builtin(__builtin_amdgcn_s_wait_asynccnt)
#define USE_ASYNC_LDS 1
#endif

// ---------------------------------------------------------------------------
// Kernel 1: per-chunk top-32 candidate selection.
// key = (|f| bits << 32) | (N-1-idx): u64 max => largest |v|, tie => smaller idx
// (matches stable argsort(-|x|) tie-breaking).
// ---------------------------------------------------------------------------
__global__ __launch_bounds__(NTHREADS)
void roc_select(const float* __restrict__ data, u64* __restrict__ cand) {
  __shared__ float raw[CHUNK];       // 16 KB
  __shared__ u64   keys[CHUNK];      // 32 KB
  __shared__ u64   red[NTHREADS];    // 2 KB

  const int tid  = threadIdx.x;
  const int b    = blockIdx.x / NBLK;
  const int c    = blockIdx.x % NBLK;
  const int base = b * RN + c * CHUNK;

  // ---- Stage CHUNK floats into LDS ----
#ifdef USE_ASYNC_LDS
  // 4 passes x 256 lanes x 16B = 16 KB, tracked by ASYNCcnt.
  #pragma unroll
  for (int p = 0; p < CHUNK / (NTHREADS * 4); ++p) {
    const int e = p * NTHREADS * 4 + tid * 4;
    __builtin_amdgcn_global_load_async_to_lds_b128(
        (__attribute__((address_space(1))) v4i*)(data + base + e),
        (__attribute__((address_space(3))) v4i*)(&raw[e]),
        0, 0);
  }
  __builtin_amdgcn_s_wait_asynccnt(0);
#else
  {
    const f4* g4 = (const f4*)(data + base);
    f4* r4 = (f4*)raw;
    for (int p = tid; p < CHUNK / 4; p += NTHREADS) r4[p] = g4[p];
  }
#endif
  __syncthreads();

  // ---- Build sortable keys ----
  for (int i = tid; i < CHUNK; i += NTHREADS) {
    const u32 a    = __float_as_uint(raw[i]) & 0x7fffffffu;
    const u32 gidx = (u32)(c * CHUNK + i);              // batch-local index
    keys[i] = ((u64)a << 32) | (u64)((u32)(RN - 1u) - gidx);
  }
  __syncthreads();

  // ---- 32 rounds of block-wide argmax, invalidating each winner ----
  for (int r = 0; r < CAND; ++r) {
    u64 m = 0;
    for (int i = tid; i < CHUNK; i += NTHREADS) {
      const u64 k = keys[i];
      m = (k > m) ? k : m;
    }
    red[tid] = m;
    __syncthreads();
    for (int s = NTHREADS / 2; s > 0; s >>= 1) {
      if (tid < s) {
        const u64 o = red[tid + s];
        if (o > red[tid]) red[tid] = o;
      }
      __syncthreads();
    }
    const u64 w = red[0];
    if (tid == 0) {
      cand[(u32)blockIdx.x * CAND + r] = w;
      if (w) {
        const u32 gidx = (u32)(RN - 1u) - (u32)(w & 0xffffffffu);
        keys[gidx - (u32)(c * CHUNK)] = 0;  // invalidate winner
      }
    }
    __syncthreads();
  }
}

// ---------------------------------------------------------------------------
// Kernel 2: per-batch, select top-31 from 64*32 candidates -> rank list.
// ---------------------------------------------------------------------------
__global__ __launch_bounds__(NTHREADS)
void roc_rank(const u64* __restrict__ cand, u32* __restrict__ ranks) {
  __shared__ u64 keys[NBLK * CAND];   // 2048 keys, 16 KB
  __shared__ u64 red[NTHREADS];

  const int tid = threadIdx.x;
  const int b   = blockIdx.x;

  for (int i = tid; i < NBLK * CAND; i += NTHREADS)
    keys[i] = cand[b * (NBLK * CAND) + i];
  __syncthreads();

  for (int r = 0; r < TOPK; ++r) {
    u64 m = 0;
    for (int i = tid; i < NBLK * CAND; i += NTHREADS) {
      const u64 k = keys[i];
      m = (k > m) ? k : m;
    }
    red[tid] = m;
    __syncthreads();
    for (int s = NTHREADS / 2; s > 0; s >>= 1) {
      if (tid < s) {
        const u64 o = red[tid + s];
        if (o > red[tid]) red[tid] = o;
      }
      __syncthreads();
    }
    const u64 w = red[0];
    if (tid == 0)
      ranks[b * RT + r] = (u32)(RN - 1u) - (u32)(w & 0xffffffffu);
    // invalidate winner (keys are unique since low bits carry the index)
    if (w) {
      for (int i = tid; i < NBLK * CAND; i += NTHREADS)
        if (keys[i] == w) keys[i] = 0;
    }
    __syncthreads();
  }
}

// ---------------------------------------------------------------------------
// Kernel 3: streaming fill of the 256 MB output. out[b,t,:] = (t==31) ? 1 : 0.
// N/4 = 65536 float4 per (b,t) slice -> t = (vec_index >> 16) & 31.
// Non-temporal b128 stores: pure write stream, keep it out of L2.
// ---------------------------------------------------------------------------
__global__ __launch_bounds__(NTHREADS)
void roc_fill(f4* __restrict__ out) {
  const u32 base = (u32)blockIdx.x * (NTHREADS * FILL_IT) + threadIdx.x;
  #pragma unroll
  for (int k = 0; k < FILL_IT; ++k) {
    const u32 i = base + (u32)k * NTHREADS;
    const u32 t = (i >> 16) & 31u;
    const float v = (t == (RT - 1)) ? 1.0f : 0.0f;
    const f4 val = {v, v, v, v};
    __builtin_nontemporal_store(val, out + i);
  }
}

// ---------------------------------------------------------------------------
// Kernel 4: fixup. For rank r<31: out[b,r,idx]=1 and out[b,31,idx]=0.
// ---------------------------------------------------------------------------
__global__ __launch_bounds__(NTHREADS)
void roc_fixup(float* __restrict__ out, const u32* __restrict__ ranks) {
  const int i = threadIdx.x;
  if (i < RB * TOPK) {
    const int b = i / TOPK;
    const int r = i % TOPK;
    const u32 idx = ranks[b * RT + r];
    if (idx < (u32)RN) {
      out[(size_t)(b * RT + r) * RN + idx]        = 1.0f;
      out[(size_t)(b * RT + (RT - 1)) * RN + idx] = 0.0f;
    }
  }
}

// ---------------------------------------------------------------------------
extern "C" void kernel_launch(void* const* d_in, const int* in_sizes, int n_in,
                              void* d_out, int out_size, void* d_ws, size_t ws_size,
                              hipStream_t stream) {
  (void)in_sizes; (void)n_in; (void)out_size; (void)ws_size;

  const float* data = (const float*)d_in[0];
  float* out = (float*)d_out;

  // Workspace layout: [0, 128KB): candidate keys; [128KB, +1KB): rank lists.
  u64* cand  = (u64*)d_ws;
  u32* ranks = (u32*)((char*)d_ws + (size_t)RB * NBLK * CAND * sizeof(u64));

  roc_select<<<RB * NBLK, NTHREADS, 0, stream>>>(data, cand);
  roc_rank<<<RB, NTHREADS, 0, stream>>>(cand, ranks);

  const int fill_blocks = (RB * RT * (RN / 4)) / (NTHREADS * FILL_IT); // 8192
  roc_fill<<<fill_blocks, NTHREADS, 0, stream>>>((f4*)out);
  roc_fixup<<<1, NTHREADS, 0, stream>>>(out, ranks);
}